// Sep_Rot_22789096473425
// MI455X (gfx1250) — compile-verified
//
#include <hip/hip_runtime.h>
#include <hip/hip_bf16.h>
#include <math.h>

// Problem constants (match reference setup_inputs).
#define BB    4
#define HH    64
#define WWID  64
#define HWPIX (HH * WWID)     // 4096
#define NVV   16
#define VLL   4
#define NSS   16
#define KS2   9

typedef __attribute__((ext_vector_type(2))) float v2f;
typedef __attribute__((ext_vector_type(8))) float v8f;
typedef __attribute__((ext_vector_type(4))) unsigned int v4u;
typedef __attribute__((ext_vector_type(8))) int v8i;
typedef __attribute__((ext_vector_type(4))) int v4i;

#if defined(__gfx1250__) && __has_builtin(__builtin_amdgcn_wmma_f32_16x16x4_f32)
#define USE_WMMA 1
#else
#define USE_WMMA 0
#endif

#if defined(__gfx1250__) && __has_builtin(__builtin_amdgcn_tensor_load_to_lds)
#define USE_TDM 1
#else
#define USE_TDM 0
#endif

#if defined(__gfx1250__) && __has_builtin(__builtin_amdgcn_global_load_async_to_lds_b128)
#define USE_ASYNC 1
#else
#define USE_ASYNC 0
#endif

#define AS1 __attribute__((address_space(1)))
#define AS3 __attribute__((address_space(3)))

// 1-D linear interp on a 16-wide axis, grid_sample align_corners=False,
// padding_mode='zeros' semantics: t = (c+1)*N/2 - 0.5, 2 taps, per-tap
// validity mask folded into the weights, clamped indices.
__device__ __forceinline__ void interp16(float c, int& i0, int& i1,
                                         float& w0, float& w1) {
  float t  = (c + 1.0f) * 8.0f - 0.5f;
  float f  = floorf(t);
  float fr = t - f;
  int   ii = (int)f;
  w0 = (ii >= 0 && ii <= 15) ? (1.0f - fr) : 0.0f;
  w1 = (ii >= -1 && ii <= 14) ? fr : 0.0f;
  i0 = min(max(ii, 0), 15);
  i1 = min(max(ii + 1, 0), 15);
}

#if USE_TDM
// One TDM descriptor: copy a contiguous 4096-float (16 KB) table into LDS.
// D# per CDNA5 ISA ch8: 2-D tensor, data_size=4B, tensor_dim0=tile_dim0=4096,
// tile_dim1=1, type=2 ("image"), workgroup_mask=0 (not in a cluster).
__device__ __forceinline__ void tdm_load_table(const float* gsrc, float* ldst) {
  const unsigned long long ga = (unsigned long long)(uintptr_t)gsrc;
  const unsigned int       la = (unsigned int)(uintptr_t)ldst;  // low 32b = LDS offset
  v4u g0;
  g0[0] = 1u;                                                   // count=1, user D#
  g0[1] = la;                                                   // lds_addr
  g0[2] = (unsigned int)ga;                                     // global_addr[31:0]
  g0[3] = (unsigned int)((ga >> 32) & 0x01FFFFFFull) | (2u << 30); // [56:32] | type=2
  v8i g1;
  g1[0] = 0x00020000;            // workgroup_mask=0, data_size=2 (4B), no pad/iter
  g1[1] = (int)(4096u << 16);    // atomic_barrier_addr=0 | tensor_dim0[15:0]=4096
  g1[2] = (int)(1u << 16);       // tensor_dim0[31:16]=0 | tensor_dim1[15:0]=1
  g1[3] = (int)(4096u << 16);    // tensor_dim1[31:16]=0 | tile_dim0=4096
  g1[4] = 1;                     // tile_dim1=1 | tile_dim2=0
  g1[5] = 4096;                  // tensor_dim0_stride[31:0]
  g1[6] = (int)(4096u << 16);    // stride0[47:32]=0 | tensor_dim1_stride[15:0]
  g1[7] = 0;
  v4i g2 = {0, 0, 0, 0}, g3 = {0, 0, 0, 0};                     // 2-D: unused
  v8i g4 = {0, 0, 0, 0, 0, 0, 0, 0};                            // clang-23 extra group
  __builtin_amdgcn_tensor_load_to_lds(g0, g1, g2, g3, g4, 0);
}
#endif

__global__ __launch_bounds__(64) void sep_rot_kernel(
    const float* __restrict__ grid1,   // [B,H,W,2]
    const float* __restrict__ grid2,   // [B,H,W,2]
    const float* __restrict__ theta,   // [B]
    const float* __restrict__ lgn,     // [B,64,H,W]
    const float* __restrict__ coefx,   // [NV,VL,VL,NS]
    const float* __restrict__ coefy,   // [NV,VL,VL,NS]
    const float* __restrict__ coefr,   // [NV,VL,VL,NS]
    float* __restrict__ out)           // [B,64,H,W]
{
  __shared__ __align__(16) float sCx[NVV * VLL * VLL * NSS];  // 16 KB
  __shared__ __align__(16) float sCr[NVV * VLL * VLL * NSS];  // 16 KB
  __shared__ __align__(16) float sMz[NVV * VLL * VLL];        // 1 KB
  __shared__ __align__(16) float sE[2][4][16][32];            // 16 KB scratch

  const int tid  = blockIdx.x * 64 + threadIdx.x;
  const int b    = tid >> 12;          // 64 blocks per batch -> uniform b
  const int pix  = tid & 4095;
  const int lane = threadIdx.x & 31;
  const int wv   = threadIdx.x >> 5;

  // ---- Stage coefx/coefr into LDS (TDM if available) ----
#if USE_TDM
  if (threadIdx.x < 32) {              // wave 0 issues both DMAs
    tdm_load_table(coefx, &sCx[0]);
    tdm_load_table(coefr, &sCr[0]);
  }
#else
  for (int i = threadIdx.x; i < NVV * VLL * VLL * NSS; i += 64) {
    sCx[i] = coefx[i];
    sCr[i] = coefr[i];
  }
#endif

  // ---- Stage coefy into the (currently dead) sE scratch via async loads ----
  float* sEf = &sE[0][0][0][0];
#if USE_ASYNC
  for (int i = threadIdx.x * 4; i < NVV * VLL * VLL * NSS; i += 64 * 4)
    __builtin_amdgcn_global_load_async_to_lds_b128(
        (AS1 v4i*)(uintptr_t)(coefy + i),
        (AS3 v4i*)(uintptr_t)(sEf + i), 0, 0);
#if __has_builtin(__builtin_amdgcn_s_wait_asynccnt)
  __builtin_amdgcn_s_wait_asynccnt(0);
#else
  asm volatile("s_wait_asynccnt 0" ::: "memory");
#endif
#endif
#if USE_TDM
  if (threadIdx.x < 32) {
#if __has_builtin(__builtin_amdgcn_s_wait_tensorcnt)
    __builtin_amdgcn_s_wait_tensorcnt(0);
#else
    asm volatile("s_wait_tensorcnt 0" ::: "memory");
#endif
  }
#endif
  __syncthreads();

  // ---- Per-batch theta interp folded into coefy -> Mz table ----
  {
    float z = theta[b] * (1.0f / 3.14159265358979323846f);
    int z0, z1; float wz0, wz1;
    interp16(z, z0, z1, wz0, wz1);
    for (int i = threadIdx.x; i < NVV * VLL * VLL; i += 64) {
#if USE_ASYNC
      sMz[i] = wz0 * sEf[i * NSS + z0] + wz1 * sEf[i * NSS + z1];
#else
      sMz[i] = wz0 * coefy[i * NSS + z0] + wz1 * coefy[i * NSS + z1];
#endif
    }
  }
  __syncthreads();   // sMz ready; sE free for reuse as WMMA spill scratch

  // ---- Per-pixel nearest warp by grid2 (jnp.round == RNE -> rintf) ----
  const float gx2 = grid2[(b * HWPIX + pix) * 2 + 0];
  const float gy2 = grid2[(b * HWPIX + pix) * 2 + 1];
  const float rx  = rintf((gx2 + 1.0f) * 32.0f - 0.5f);
  const float ry  = rintf((gy2 + 1.0f) * 32.0f - 0.5f);
  const bool  v0  = (rx >= 0.0f && rx <= 63.0f && ry >= 0.0f && ry <= 63.0f);
  const int   ix  = min(max((int)rx, 0), 63);
  const int   iy  = min(max((int)ry, 0), 63);

  // ---- Per-tap params, fully unrolled so arrays live in registers ----
  int   offT[KS2];
  bool  mT[KS2];
  int   cx0T[KS2], cx1T[KS2], ny0T[KS2], ny1T[KS2];
  float wx0T[KS2], wx1T[KS2], wy0T[KS2], wy1T[KS2];
#pragma unroll
  for (int t = 0; t < KS2; ++t) {
    const int di = t / 3 - 1, dj = t % 3 - 1;
    const int sy = iy + di, sx = ix + dj;
    const bool m = v0 && sy >= 0 && sy < HH && sx >= 0 && sx < WWID;
    const int syc = min(max(sy, 0), 63), sxc = min(max(sx, 0), 63);
    mT[t]   = m;
    offT[t] = b * (64 * HWPIX) + syc * WWID + sxc;   // +channel*4096 later
    const int gi = ((b * HH + syc) * WWID + sxc) * 2;
    const float g1x = m ? grid1[gi + 0] : 0.0f;
    const float g1y = m ? grid1[gi + 1] : 0.0f;
    interp16(gx2 - g1x, cx0T[t], cx1T[t], wx0T[t], wx1T[t]);  // x -> coefr axis
    interp16(gy2 - g1y, ny0T[t], ny1T[t], wy0T[t], wy1T[t]);  // y -> coefx axis
  }

  const int hi = lane >> 4;   // half-wave (K split / row split)
  const int lo = lane & 15;

  for (int v = 0; v < NVV; ++v) {
    float a4[4] = {0.f, 0.f, 0.f, 0.f};
#pragma unroll
    for (int t = 0; t < KS2; ++t) {
      const int  base = offT[t] + (v * 4) * HWPIX;
      const bool m    = mT[t];
      const float lq0 = m ? lgn[base + 0 * HWPIX] : 0.0f;
      const float lq1 = m ? lgn[base + 1 * HWPIX] : 0.0f;
      const float lq2 = m ? lgn[base + 2 * HWPIX] : 0.0f;
      const float lq3 = m ? lgn[base + 3 * HWPIX] : 0.0f;
      float u[4];

#if USE_WMMA
      // E[b2, o, s] = sum_q coefr[v,b2,q,o] * lgn_s[v,q]
      // A (16x4): lane -> row o = lo, K = 2*hi + {0,1} = q. Shared coefr.
      float ak0[4], ak1[4];
#pragma unroll
      for (int mt = 0; mt < 4; ++mt) {
        ak0[mt] = sCr[((v * 4 + mt) * 4 + 2 * hi + 0) * 16 + lo];
        ak1[mt] = sCr[((v * 4 + mt) * 4 + 2 * hi + 1) * 16 + lo];
      }
#pragma unroll
      for (int g = 0; g < 2; ++g) {          // 16-sample column groups
        const int src = g * 16 + lo;
        const float t0 = __shfl(lq0, src);
        const float t1 = __shfl(lq1, src);
        const float t2 = __shfl(lq2, src);
        const float t3 = __shfl(lq3, src);
        v2f bm;                               // B (4x16): col = lo, K split @16
        bm.x = hi ? t2 : t0;
        bm.y = hi ? t3 : t1;
#pragma unroll
        for (int mt = 0; mt < 4; ++mt) {
          v2f am; am.x = ak0[mt]; am.y = ak1[mt];
          v8f c = {0.f, 0.f, 0.f, 0.f, 0.f, 0.f, 0.f, 0.f};
          c = __builtin_amdgcn_wmma_f32_16x16x4_f32(
              false, am, false, bm, (short)0, c, false, false);
          // C layout: VGPR r holds rows r (lanes 0-15) / r+8 (lanes 16-31).
#pragma unroll
          for (int r = 0; r < 8; ++r)
            sE[wv][mt][r + 8 * hi][g * 16 + lo] = c[r];
        }
      }
      // Each lane = one sample = one E column; 2-tap x-interp over o.
#pragma unroll
      for (int b2 = 0; b2 < 4; ++b2)
        u[b2] = wx0T[t] * sE[wv][b2][cx0T[t]][lane] +
                wx1T[t] * sE[wv][b2][cx1T[t]][lane];
#else
      // Scalar fallback: interp coefr then 4x4 matvec.
#pragma unroll
      for (int b2 = 0; b2 < 4; ++b2) {
        float s = 0.f;
#pragma unroll
        for (int q = 0; q < 4; ++q) {
          const int rb = ((v * 4 + b2) * 4 + q) * 16;
          const float rq = wx0T[t] * sCr[rb + cx0T[t]] +
                           wx1T[t] * sCr[rb + cx1T[t]];
          const float lq = (q == 0) ? lq0 : (q == 1) ? lq1 : (q == 2) ? lq2 : lq3;
          s += rq * lq;
        }
        u[b2] = s;
      }
#endif
      // w[a] = Mz(v) . u   (coefy interp'd at theta, per-batch constant)
      float w4[4];
#pragma unroll
      for (int a = 0; a < 4; ++a)
        w4[a] = sMz[v * 16 + a * 4 + 0] * u[0] + sMz[v * 16 + a * 4 + 1] * u[1] +
                sMz[v * 16 + a * 4 + 2] * u[2] + sMz[v * 16 + a * 4 + 3] * u[3];
      // pred[v,p] += Ax(y) . w   (coefx interp'd at y)
#pragma unroll
      for (int p = 0; p < 4; ++p) {
        float s = 0.f;
#pragma unroll
        for (int a = 0; a < 4; ++a) {
          const int cb = ((v * 4 + p) * 4 + a) * 16;
          const float axv = wy0T[t] * sCx[cb + ny0T[t]] +
                            wy1T[t] * sCx[cb + ny1T[t]];
          s += axv * w4[a];
        }
        a4[p] += s;
      }
    }  // taps
#pragma unroll
    for (int p = 0; p < 4; ++p)
      out[((b * NVV + v) * VLL + p) * HWPIX + pix] = a4[p];
  }  // v
}

extern "C" void kernel_launch(void* const* d_in, const int* in_sizes, int n_in,
                              void* d_out, int out_size, void* d_ws, size_t ws_size,
                              hipStream_t stream) {
  (void)in_sizes; (void)n_in; (void)d_ws; (void)ws_size; (void)out_size;
  const float* grid1 = (const float*)d_in[0];
  const float* grid2 = (const float*)d_in[1];
  const float* theta = (const float*)d_in[2];
  const float* lgn   = (const float*)d_in[3];
  const float* coefx = (const float*)d_in[4];
  const float* coefy = (const float*)d_in[5];
  const float* coefr = (const float*)d_in[6];
  float* out = (float*)d_out;
  // 16384 pixels total, 64 threads (2 wave32) per block, one batch per block.
  sep_rot_kernel<<<(BB * HWPIX) / 64, 64, 0, stream>>>(
      grid1, grid2, theta, lgn, coefx, coefy, coefr, out);
}